// Leafnet_57543971831919
// MI455X (gfx1250) — compile-verified
//
#include <hip/hip_runtime.h>
#include <hip/hip_bf16.h>
#include <math.h>

// ---------------------------------------------------------------------------
// Problem constants (from the reference)
// ---------------------------------------------------------------------------
#define BATCH   32
#define CHAN    3
#define HH      257
#define WW      257
#define MASK    17
#define STRIDE  4
#define NWIN    61                     // windows per axis
#define BC      (BATCH * CHAN)         // 96
#define NWINDOWS (BC * NWIN * NWIN)    // 357216
#define WELEMS  (MASK * MASK)          // 289

// flat output layout: sw | s | q  (all float32)
#define SW_ELEMS  ((size_t)NWINDOWS * WELEMS)            // 103,235,424
#define S_ELEMS   ((size_t)BATCH * 4 * CHAN * NWIN * NWIN) // 1,428,864
#define S_OFF     SW_ELEMS
#define Q_OFF     (SW_ELEMS + S_ELEMS)                   // 104,664,288

// workspace: 4 column-stat planes of (BC, 257, 61) floats = ~24.1 MB
#define CS_ELEMS  ((size_t)BC * HH * NWIN)               // 1,504,992

typedef __attribute__((ext_vector_type(2))) float v2f;
typedef __attribute__((ext_vector_type(8))) float v8f;

// ---------------------------------------------------------------------------
// Kernel 1 (bandwidth-dominant): write sw and q for every window element.
// One wave per window; non-temporal b32 stores (write-once streams) so the
// 832 MB of output does not evict the tiny, heavily-reused input from L2.
// ---------------------------------------------------------------------------
__global__ void swq_kernel(const float* __restrict__ x, float* __restrict__ out) {
    const int wave = (int)((blockIdx.x * blockDim.x + threadIdx.x) >> 5);
    const int lane = (int)(threadIdx.x & 31);
    if (wave >= NWINDOWS) return;

    const int wx = wave % NWIN;
    const int t1 = wave / NWIN;
    const int wy = t1 % NWIN;
    const int bc = t1 / NWIN;

    const float* __restrict__ xb =
        x + ((size_t)bc * HH + (size_t)wy * STRIDE) * WW + (size_t)wx * STRIDE;

    float* __restrict__ swp = out + (size_t)wave * WELEMS;
    float* __restrict__ qp  = out + Q_OFF + (size_t)wave * WELEMS;

#pragma unroll
    for (int t = 0; t < 10; ++t) {               // ceil(289/32) iterations
        const int e = lane + 32 * t;
        if (e < WELEMS) {
            const int i = e / MASK;
            const int j = e - i * MASK;
            const float v = xb[i * WW + j];
            // digitize(v, linspace(0,1,65)) - 1 == floor(v*64) for v in [0,1)
            const float qv = floorf(v * 64.0f);
            __builtin_nontemporal_store(v,  swp + e);
            __builtin_nontemporal_store(qv, qp  + e);
        }
    }
}

// ---------------------------------------------------------------------------
// Kernel 2: horizontal pass. For every image row h and window column wx,
// reduce the 17-wide horizontal window -> colsum/colsumsq/colmax/colmin.
// One block per (bc,h) row; thread t handles window column wx=t.
// ---------------------------------------------------------------------------
__global__ void colstats_kernel(const float* __restrict__ x,
                                float* __restrict__ cs, float* __restrict__ cq,
                                float* __restrict__ cmx, float* __restrict__ cmn) {
    const int row = (int)blockIdx.x;             // bc*257 + h
    const int t   = (int)threadIdx.x;
    if (t >= NWIN) return;

    const float* __restrict__ xr = x + (size_t)row * WW;
    const int base = t * STRIDE;

    float s = 0.0f, q = 0.0f;
    float mx = -3.402823466e38f, mn = 3.402823466e38f;
#pragma unroll
    for (int j = 0; j < MASK; ++j) {
        const float v = xr[base + j];
        s += v; q += v * v;
        mx = fmaxf(mx, v); mn = fminf(mn, v);
    }
    const size_t o = (size_t)row * NWIN + t;
    cs[o] = s; cq[o] = q; cmx[o] = mx; cmn[o] = mn;
}

// ---------------------------------------------------------------------------
// Kernel 3: vertical pass as a banded GEMM on the matrix core.
//   winsum  (61x61) = S(61x257) . colsum (257x61)
//   winsumsq(61x61) = S(61x257) . colsumsq(257x61)
// where S[m][h] = 1 iff 4m <= h < 4m+17.  Each wave owns one 16x16 output
// tile; only 20 K-chunks of 4 are non-zero for a given row tile.
// V_WMMA_F32_16X16X4_F32 operand layouts (ISA 7.12.2):
//   A 16x4: v2f/lane, M = lane&15, K = v + (lane<16 ? 0 : 2)
//   B 4x16: v2f/lane, N = lane&15, K = v + (lane<16 ? 0 : 2)   (symmetric)
//   C/D 16x16: v8f/lane, M = v + (lane<16 ? 0 : 8), N = lane&15
// Max/min finish with 17-deep VALU reductions, then the four `s` stats.
// ---------------------------------------------------------------------------
__global__ void winstats_wmma_kernel(const float* __restrict__ cs,
                                     const float* __restrict__ cq,
                                     const float* __restrict__ cmx,
                                     const float* __restrict__ cmn,
                                     float* __restrict__ out) {
    const int wave = (int)((blockIdx.x * blockDim.x + threadIdx.x) >> 5);
    const int lane = (int)(threadIdx.x & 31);
    if (wave >= BC * 16) return;

    const int tile = wave & 15;                  // 4x4 tiles of the 61x61 grid
    const int bc   = wave >> 4;
    const int m0 = (tile >> 2) * 16;
    const int n0 = (tile & 3) * 16;

    const float* __restrict__ csB = cs  + (size_t)bc * HH * NWIN;
    const float* __restrict__ cqB = cq  + (size_t)bc * HH * NWIN;
    const float* __restrict__ mxB = cmx + (size_t)bc * HH * NWIN;
    const float* __restrict__ mnB = cmn + (size_t)bc * HH * NWIN;

    const int halfK = (lane < 16) ? 0 : 2;
    const int mA = m0 + (lane & 15);             // A-matrix row this lane feeds
    const int nB = min(n0 + (lane & 15), NWIN - 1); // clamped B column (OOB masked later)
    const int k0 = m0 * STRIDE;                  // first non-zero K for this row tile

    v8f accS = {};                               // window sums
    v8f accQ = {};                               // window sums of squares

    for (int kk = 0; kk < 80; kk += 4) {         // 20 K-steps cover k0..k0+79
        v2f a, bS, bQ;
#pragma unroll
        for (int v = 0; v < 2; ++v) {
            const int k = k0 + kk + halfK + v;
            a[v] = (k >= STRIDE * mA && k < STRIDE * mA + MASK) ? 1.0f : 0.0f;
            const int kc = (k > HH - 1) ? (HH - 1) : k;   // clamp address; A==0 there
            bS[v] = csB[(size_t)kc * NWIN + nB];
            bQ[v] = cqB[(size_t)kc * NWIN + nB];
        }
        accS = __builtin_amdgcn_wmma_f32_16x16x4_f32(
            false, a, false, bS, (short)0, accS, false, false);
        accQ = __builtin_amdgcn_wmma_f32_16x16x4_f32(
            false, a, false, bQ, (short)0, accQ, false, false);
    }

    // Finish: max/min reductions + the four normalized stats, masked store.
    const int mOff = (lane < 16) ? 0 : 8;
    const int n = n0 + (lane & 15);
    const int b = bc / CHAN;
    const int c = bc % CHAN;
    const float invStd = 4.0f;                   // 1 / G_STD
    const float invCnt = 1.0f / (float)WELEMS;

#pragma unroll
    for (int v = 0; v < 8; ++v) {
        const int m = m0 + v + mOff;
        if (m < NWIN && n < NWIN) {
            const int h0 = m * STRIDE;
            float mx = -3.402823466e38f, mn = 3.402823466e38f;
#pragma unroll
            for (int i = 0; i < MASK; ++i) {
                mx = fmaxf(mx, mxB[(size_t)(h0 + i) * NWIN + n]);
                mn = fminf(mn, mnB[(size_t)(h0 + i) * NWIN + n]);
            }
            const float mean = accS[v] * invCnt;
            const float msq  = accQ[v] * invCnt;
            const float sd   = sqrtf(fmaxf(msq - mean * mean, 0.0f));

            // s layout: (B, 4C, 61, 61), channel = stat*C + c
            const size_t sb = S_OFF + ((((size_t)b * 4 * CHAN) * NWIN + m) * NWIN + n);
            const size_t chStride = (size_t)NWIN * NWIN;   // stride for +1 channel
            out[sb + (size_t)(0 * CHAN + c) * chStride] = (mx - 0.5f) * invStd;
            out[sb + (size_t)(1 * CHAN + c) * chStride] = sd * invStd;
            out[sb + (size_t)(2 * CHAN + c) * chStride] = (mx - mean) * invStd;
            out[sb + (size_t)(3 * CHAN + c) * chStride] = (mean - mn) * invStd;
        }
    }
}

// ---------------------------------------------------------------------------
// Launcher
// ---------------------------------------------------------------------------
extern "C" void kernel_launch(void* const* d_in, const int* in_sizes, int n_in,
                              void* d_out, int out_size, void* d_ws, size_t ws_size,
                              hipStream_t stream) {
    (void)in_sizes; (void)n_in; (void)out_size; (void)ws_size;
    const float* x = (const float*)d_in[0];
    // d_in[1] = bins: known to be linspace(0,1,65); quantization folded into floor(v*64).
    float* out = (float*)d_out;
    float* ws  = (float*)d_ws;

    float* cs  = ws;
    float* cq  = ws + CS_ELEMS;
    float* cmx = ws + 2 * CS_ELEMS;
    float* cmn = ws + 3 * CS_ELEMS;

    // Dominant streaming kernel: 1 wave/window, 8 waves/block.
    swq_kernel<<<NWINDOWS / 8, 256, 0, stream>>>(x, out);

    // Horizontal stats pass: one block per (bc, h) row.
    colstats_kernel<<<BC * HH, 64, 0, stream>>>(x, cs, cq, cmx, cmn);

    // Vertical pass on the matrix core: BC*16 tiles, 4 waves/block.
    winstats_wmma_kernel<<<(BC * 16) / 4, 128, 0, stream>>>(cs, cq, cmx, cmn, out);
}